// SphericalGaborLayer_51290499448968
// MI455X (gfx1250) — compile-verified
//
#include <hip/hip_runtime.h>
#include <hip/hip_bf16.h>
#include <math.h>

// ---------------------------------------------------------------------------
// SphericalGaborLayer for MI455X (gfx1250, wave32).
//
// Roofline: output is 131072x512 fp32 = 256MB -> HBM store floor ~11us at
// 23.3 TB/s. Per-filter trig/exp is precomputed (F=512 only). The rotation
// contraction (K=4 matvec producing x and z+1+eps per (n,f)) is mapped onto
// V_WMMA_F32_16X16X4_F32: two WMMAs yield a 16-point x 16-filter tile of x
// and z in matching lane/VGPR positions, so the transcendental tail
// (v_rcp_f32, v_cos_f32, v_exp_f32) runs with zero cross-lane traffic.
// Constants are pre-folded: cos scale includes 1/(2pi) (v_cos convention),
// exp scale includes log2(e) (v_exp is exp2), and the 1+eps bias rides in
// the 4th row of the B(z) matrix via the A-column of ones.
// ---------------------------------------------------------------------------

typedef __attribute__((ext_vector_type(2))) float v2f;
typedef __attribute__((ext_vector_type(8))) float v8f;

#define EPSV 1e-6f

constexpr int NPTS = 131072;
constexpr int NFLT = 512;
constexpr int PTS_PER_BLOCK = 256;   // 16 point-tiles of 16 per wave loop
constexpr float TWO_PI = 6.28318530717958647692f;
constexpr float LOG2E  = 1.44269504088896340736f;

// ---------------------------------------------------------------------------
// Phase 1: per-filter constants (rows 0 and 2 of R = Rz(g) Rx(b) Rz(a),
// plus folded cos/exp scales). SoA layout in workspace, 8 arrays of NFLT.
// ---------------------------------------------------------------------------
__global__ void gabor_filter_setup(const float* __restrict__ dilate,
                                   const float* __restrict__ u,
                                   const float* __restrict__ v,
                                   const float* __restrict__ w,
                                   float* __restrict__ ws) {
    int f = blockIdx.x * blockDim.x + threadIdx.x;
    if (f >= NFLT) return;

    float alpha = TWO_PI * u[f];
    float cv    = 2.0f * v[f] - 1.0f;
    cv          = fminf(fmaxf(cv, -1.0f + EPSV), 1.0f - EPSV);
    float beta  = acosf(cv);
    float gamma = TWO_PI * w[f];

    float sa = sinf(alpha), ca = cosf(alpha);
    float sb = sinf(beta),  cb = cosf(beta);
    float sg = sinf(gamma), cg = cosf(gamma);

    // R = Rz(gamma) * Rx(beta) * Rz(alpha); only rows 0 and 2 are used.
    float r00 =  cg * ca - sg * cb * sa;
    float r01 = -cg * sa - sg * cb * ca;
    float r02 =  sg * sb;
    float r20 =  sb * sa;
    float r21 =  sb * ca;
    float r22 =  cb;

    float d  = expf(dilate[f]);
    // cos(OMEGA * 2*d*x*inv) = v_cos( (60*d/(2pi)) * x * inv )
    float k1 = (60.0f / TWO_PI) * d;
    // exp(-SIGMA^2 * 4*d^2*(1-z)*inv) = v_exp2( (-400*d^2*log2e) * (1-z) * inv )
    float k2 = -400.0f * d * d * LOG2E;

    ws[0 * NFLT + f] = r00;
    ws[1 * NFLT + f] = r01;
    ws[2 * NFLT + f] = r02;
    ws[3 * NFLT + f] = r20;
    ws[4 * NFLT + f] = r21;
    ws[5 * NFLT + f] = r22;
    ws[6 * NFLT + f] = k1;
    ws[7 * NFLT + f] = k2;
}

// ---------------------------------------------------------------------------
// Phase 2: WMMA-tiled evaluation.
//   block = 256 threads = 8 waves; gridDim = (NPTS/256, NFLT/(16*8)).
//   Each wave owns one 16-filter tile and loops 16 point-tiles of 16 points.
//   Per tile: A(16x4) = [px,py,pz,1], D = A x B  via v_wmma_f32_16x16x4_f32.
//   B(x): rows {r00,r01,r02,0};  B(z): rows {r20,r21,r22,1+eps}  (bias fold).
//   B fragment layout assumption (mirrors documented A layout): v0 holds
//   rows {K0 | K2} across lane halves, v1 holds rows {K1 | K3}.
// ---------------------------------------------------------------------------
__global__ void __launch_bounds__(256)
gabor_wmma_eval(const float* __restrict__ pts,
                const float* __restrict__ ws,
                float* __restrict__ out) {
    const int lane = threadIdx.x & 31;
    const int wave = threadIdx.x >> 5;
    const int ft   = blockIdx.y * 8 + wave;   // filter tile id, 0..31
    const int f0   = ft * 16;
    const int nc   = lane & 15;               // N column within tile
    const bool hi  = lane >= 16;
    const int f    = f0 + nc;                 // this lane's filter

    const float* bx0 = ws + 0 * NFLT;
    const float* bx1 = ws + 1 * NFLT;
    const float* bx2 = ws + 2 * NFLT;
    const float* bz0 = ws + 3 * NFLT;
    const float* bz1 = ws + 4 * NFLT;
    const float* bz2 = ws + 5 * NFLT;
    const float* k1a = ws + 6 * NFLT;
    const float* k2a = ws + 7 * NFLT;

    // B fragments (loaded once per wave, reused for 256 points).
    v2f bx, bz;
    bx.x = hi ? bx2[f] : bx0[f];
    bx.y = hi ? 0.0f   : bx1[f];
    bz.x = hi ? bz2[f] : bz0[f];
    bz.y = hi ? (1.0f + EPSV) : bz1[f];
    const float k1 = k1a[f];
    const float k2 = k2a[f];

    const int chunk0 = blockIdx.x * PTS_PER_BLOCK;
    const int plo    = lane & 15;             // point row within tile half

    for (int t = 0; t < PTS_PER_BLOCK / 16; ++t) {
        const int n0   = chunk0 + t * 16;
        const int pidx = n0 + plo;

        // A fragment: lanes 0-15 -> {px,py} (K0,K1); lanes 16-31 -> {pz,1} (K2,K3).
        v2f a;
        a.x = pts[3 * pidx + (hi ? 2 : 0)];
        a.y = hi ? 1.0f : pts[3 * pidx + 1];

        v8f zero = {};
        // D_x[m, f] = row0 . p ; D_z[m, f] = row2 . p + (1 + eps)
        v8f xv = __builtin_amdgcn_wmma_f32_16x16x4_f32(
            false, a, false, bx, (short)0, zero, false, false);
        v8f zv = __builtin_amdgcn_wmma_f32_16x16x4_f32(
            false, a, false, bz, (short)0, zero, false, false);

        float* orow = out + (size_t)(n0 + (hi ? 8 : 0)) * NFLT + f;
        #pragma unroll
        for (int r = 0; r < 8; ++r) {
            float x   = xv[r];
            float zp  = zv[r];                         // 1 + eps + z
            float inv = __builtin_amdgcn_rcpf(zp);     // v_rcp_f32
            float c   = __builtin_amdgcn_cosf(k1 * x * inv);          // v_cos_f32
            float e   = __builtin_amdgcn_exp2f(k2 * ((2.0f + EPSV) - zp) * inv); // v_exp_f32
            orow[(size_t)r * NFLT] = c * e;            // coalesced 64B/half-wave
        }
    }
}

// ---------------------------------------------------------------------------
extern "C" void kernel_launch(void* const* d_in, const int* in_sizes, int n_in,
                              void* d_out, int out_size, void* d_ws, size_t ws_size,
                              hipStream_t stream) {
    const float* inp    = (const float*)d_in[0];  // (N, 3)
    const float* dilate = (const float*)d_in[1];  // (1, F)
    const float* u      = (const float*)d_in[2];  // (F,)
    const float* v      = (const float*)d_in[3];  // (F,)
    const float* w      = (const float*)d_in[4];  // (F,)
    float*       out    = (float*)d_out;          // (N, F)
    float*       ws     = (float*)d_ws;           // 8 * NFLT floats = 16 KB

    gabor_filter_setup<<<dim3(NFLT / 256), dim3(256), 0, stream>>>(dilate, u, v, w, ws);

    dim3 grid(NPTS / PTS_PER_BLOCK, NFLT / (16 * 8));  // (512, 4)
    gabor_wmma_eval<<<grid, dim3(256), 0, stream>>>(inp, ws, out);
}